// ResidualBidirectionalMambaLayer_84293028151645
// MI455X (gfx1250) — compile-verified
//
#include <hip/hip_runtime.h>
#include <hip/hip_bf16.h>

#define DIMX    768
#define D_INNER 1536
#define D_STATE 16
#define D_CONV  4
#define DT_RANK 48
#define LSEQ    1024
#define NBATCH  2
#define MROWS   (NBATCH * LSEQ)   // 2048

typedef __attribute__((ext_vector_type(2))) float v2f;
typedef __attribute__((ext_vector_type(4))) float v4f;
typedef __attribute__((ext_vector_type(8))) float v8f;

__device__ __forceinline__ v8f wmma4(v2f a, v2f b, v8f c) {
    return __builtin_amdgcn_wmma_f32_16x16x4_f32(false, a, false, b,
                                                 (short)0, c, false, false);
}

// ---------------------------------------------------------------------------
// f32 GEMM on the CDNA5 WMMA pipe:  C[M,N] = A[M,K] @ W[N,K]^T
// A row-major (stride lda); W row-major [N,K] (stride K); C stride N.
// One wave -> 32(M) x 64(N) strip; K processed in chunks of 16 with a
// K-permuted fragment mapping: lane-half h covers physical k in
// [k0+8h, k0+8h+7], so each lane's data for 4 WMMA steps is 8 contiguous
// floats = two b128 loads.  12 b128 loads feed 32 WMMAs per chunk, with the
// next chunk's fragments prefetched before the current WMMAs issue.
// Partial N tiles: B row index clamped (loads in-bounds, EXEC stays full),
// stores guarded by wave-uniform valid[j].
// rev!=0: A rows read time-reversed within each Lseq batch (backward dir).
// ---------------------------------------------------------------------------
__global__ __launch_bounds__(128)
void gemm_f32_wmma(const float* __restrict__ A, int lda,
                   const float* __restrict__ W,
                   float* __restrict__ C,
                   int M, int N, int K, int rev, int Lseq)
{
    const int lane = threadIdx.x & 31;
    const int wave = threadIdx.x >> 5;
    const int half = lane >> 4;     // selects k-range [8h, 8h+7] within chunk
    const int idx  = lane & 15;     // M index (A) / N index (B, D col)

    const int m0 = (blockIdx.y * 4 + wave) * 32;
    const int n0 = blockIdx.x * 64;
    if (m0 >= M) return;

    // Per-lane A rows (optionally time-reversed), hoisted out of the K loop.
    int mA = m0 + idx, mB = m0 + 16 + idx;
    if (rev) {
        int b = mA / Lseq, l = mA - b * Lseq;
        mA = b * Lseq + (Lseq - 1 - l);
        b = mB / Lseq; l = mB - b * Lseq;
        mB = b * Lseq + (Lseq - 1 - l);
    }
    const float* __restrict__ ar0 = A + (size_t)mA * lda + 8 * half;
    const float* __restrict__ ar1 = A + (size_t)mB * lda + 8 * half;

    bool valid[4];
    const float* __restrict__ wr[4];
    #pragma unroll
    for (int j = 0; j < 4; ++j) {
        const int nj = n0 + j * 16;
        valid[j] = (nj + 16 <= N);                 // wave-uniform
        int nr = nj + idx;
        if (nr >= N) nr = N - 1;                   // clamp: loads stay in-bounds
        wr[j] = W + (size_t)nr * K + 8 * half;
    }

    v8f acc0[4] = {}, acc1[4] = {};

    auto loadc = [&](int k0, v4f (&A0)[2], v4f (&A1)[2], v4f (&Bf)[4][2]) {
        A0[0] = *(const v4f*)(ar0 + k0);
        A0[1] = *(const v4f*)(ar0 + k0 + 4);
        A1[0] = *(const v4f*)(ar1 + k0);
        A1[1] = *(const v4f*)(ar1 + k0 + 4);
        #pragma unroll
        for (int j = 0; j < 4; ++j) {
            Bf[j][0] = *(const v4f*)(wr[j] + k0);
            Bf[j][1] = *(const v4f*)(wr[j] + k0 + 4);
        }
    };
    auto compute = [&](v4f (&A0)[2], v4f (&A1)[2], v4f (&Bf)[4][2]) {
        #pragma unroll
        for (int i = 0; i < 4; ++i) {              // 4 WMMA K-steps per chunk
            const int q = i >> 1, e = 2 * (i & 1);
            v2f fa0; fa0.x = A0[q][e]; fa0.y = A0[q][e + 1];
            v2f fa1; fa1.x = A1[q][e]; fa1.y = A1[q][e + 1];
            #pragma unroll
            for (int j = 0; j < 4; ++j) {
                v2f fb; fb.x = Bf[j][q][e]; fb.y = Bf[j][q][e + 1];
                acc0[j] = wmma4(fa0, fb, acc0[j]);
                acc1[j] = wmma4(fa1, fb, acc1[j]);
            }
        }
    };

    v4f ca0[2], ca1[2], cb[4][2];
    loadc(0, ca0, ca1, cb);
    int k0 = 0;
    for (; k0 + 16 < K; k0 += 16) {
        v4f na0[2], na1[2], nb[4][2];
        loadc(k0 + 16, na0, na1, nb);
        compute(ca0, ca1, cb);
        #pragma unroll
        for (int q = 0; q < 2; ++q) { ca0[q] = na0[q]; ca1[q] = na1[q]; }
        #pragma unroll
        for (int j = 0; j < 4; ++j) { cb[j][0] = nb[j][0]; cb[j][1] = nb[j][1]; }
    }
    compute(ca0, ca1, cb);

    #pragma unroll
    for (int j = 0; j < 4; ++j) {
        if (valid[j]) {
            float* __restrict__ crow = C + n0 + j * 16 + idx;
            #pragma unroll
            for (int v = 0; v < 8; ++v) {
                crow[(size_t)(m0 + v + 8 * half) * N]      = acc0[j][v];
                crow[(size_t)(m0 + 16 + v + 8 * half) * N] = acc1[j][v];
            }
        }
    }
}

__device__ __forceinline__ float silu_f(float v) {
    return v / (1.f + __expf(-v));
}

// Depthwise causal conv (width 4) + SiLU.  xz holds [M, 3072]; xp = cols 0..1535.
__global__ void conv_silu_kernel(const float* __restrict__ xz,
                                 const float* __restrict__ cw,
                                 const float* __restrict__ cb,
                                 float* __restrict__ xc, int total)
{
    int gid = blockIdx.x * blockDim.x + threadIdx.x;
    if (gid >= total) return;
    int d = gid % D_INNER;
    int t = gid / D_INNER;          // t = b*L + l
    int l = t % LSEQ;
    float s = cb[d];
    #pragma unroll
    for (int k = 0; k < D_CONV; ++k) {
        int lp = l - (D_CONV - 1) + k;
        if (lp >= 0)
            s += cw[d * D_CONV + k] * xz[(size_t)(t - l + lp) * (2 * D_INNER) + d];
    }
    xc[(size_t)t * D_INNER + d] = silu_f(s);
}

// Selective-scan: one thread per (batch, channel); 16-state recurrence in regs.
// B/C vectors (x_dbl cols 48..79) are identical across all channels of a step,
// so those loads broadcast out of WGP$/L2.
__global__ void scan_kernel(const float* __restrict__ xz,     // for z (cols 1536..3071)
                            const float* __restrict__ xc,
                            const float* __restrict__ xdbl,   // [M, 80]
                            const float* __restrict__ dtlin,  // [M, 1536]
                            const float* __restrict__ b_dt,
                            const float* __restrict__ A_log,
                            const float* __restrict__ Dp,
                            float* __restrict__ y)
{
    int gid = blockIdx.x * blockDim.x + threadIdx.x;
    if (gid >= NBATCH * D_INNER) return;
    int d = gid % D_INNER;
    int b = gid / D_INNER;

    float Arow[D_STATE];
    #pragma unroll
    for (int s = 0; s < D_STATE; ++s)
        Arow[s] = -__expf(A_log[d * D_STATE + s]);

    const float dtb = b_dt[d];
    const float Dc  = Dp[d];
    float h[D_STATE] = {};

    for (int l = 0; l < LSEQ; ++l) {
        size_t row = (size_t)b * LSEQ + l;
        float v  = dtlin[row * D_INNER + d] + dtb;
        float dt = (v > 20.f) ? v : log1pf(__expf(v));
        float xcv = xc[row * D_INNER + d];
        float dx  = dt * xcv;
        const float* __restrict__ bc = xdbl + row * (DT_RANK + 2 * D_STATE);
        float acc = 0.f;
        #pragma unroll
        for (int s = 0; s < D_STATE; ++s) {
            float dA = __expf(dt * Arow[s]);
            h[s] = dA * h[s] + dx * bc[DT_RANK + s];
            acc += h[s] * bc[DT_RANK + D_STATE + s];
        }
        float zv = xz[row * (2 * D_INNER) + D_INNER + d];
        y[row * D_INNER + d] = (acc + xcv * Dc) * silu_f(zv);
    }
}

// out = x + y_fwd + flip_L(y_bwd)
__global__ void combine_kernel(const float* __restrict__ x,
                               const float* __restrict__ yfw,
                               const float* __restrict__ ybw,
                               float* __restrict__ out, int total)
{
    int gid = blockIdx.x * blockDim.x + threadIdx.x;
    if (gid >= total) return;
    int n = gid % DIMX;
    int t = gid / DIMX;
    int l = t % LSEQ;
    int b = t / LSEQ;
    size_t rrev = (size_t)(b * LSEQ + (LSEQ - 1 - l)) * DIMX + n;
    out[gid] = x[gid] + yfw[gid] + ybw[rrev];
}

extern "C" void kernel_launch(void* const* d_in, const int* in_sizes, int n_in,
                              void* d_out, int out_size, void* d_ws, size_t ws_size,
                              hipStream_t stream) {
    (void)in_sizes; (void)n_in; (void)out_size; (void)ws_size;
    const float* x = (const float*)d_in[0];

    float* ws     = (float*)d_ws;
    float* xz     = ws;                                  // [M, 3072]
    float* xc     = xz    + (size_t)MROWS * 2 * D_INNER; // [M, 1536]
    float* xdbl   = xc    + (size_t)MROWS * D_INNER;     // [M, 80]
    float* dtlin  = xdbl  + (size_t)MROWS * (DT_RANK + 2 * D_STATE); // [M, 1536]
    float* ybuf   = dtlin + (size_t)MROWS * D_INNER;     // [M, 1536]
    float* yo_fw  = ybuf  + (size_t)MROWS * D_INNER;     // [M, 768]
    float* yo_bw  = yo_fw + (size_t)MROWS * DIMX;        // [M, 768]

    const dim3 gblk(128);
    const int  gy = MROWS / 128;   // 16 block rows: 4 waves x 32 M-rows

    for (int dir = 0; dir < 2; ++dir) {
        const float* W_in   = (const float*)d_in[1 + 9 * dir];
        const float* conv_w = (const float*)d_in[2 + 9 * dir];
        const float* conv_b = (const float*)d_in[3 + 9 * dir];
        const float* W_x    = (const float*)d_in[4 + 9 * dir];
        const float* W_dt   = (const float*)d_in[5 + 9 * dir];
        const float* b_dt   = (const float*)d_in[6 + 9 * dir];
        const float* A_log  = (const float*)d_in[7 + 9 * dir];
        const float* Dp     = (const float*)d_in[8 + 9 * dir];
        const float* W_out  = (const float*)d_in[9 + 9 * dir];
        float* yout = dir ? yo_bw : yo_fw;

        // xz = x(_rev) @ W_in^T   [2048 x 3072, K=768]
        gemm_f32_wmma<<<dim3((2 * D_INNER + 63) / 64, gy), gblk, 0, stream>>>(
            x, DIMX, W_in, xz, MROWS, 2 * D_INNER, DIMX, dir, LSEQ);

        int tconv = MROWS * D_INNER;
        conv_silu_kernel<<<(tconv + 255) / 256, 256, 0, stream>>>(
            xz, conv_w, conv_b, xc, tconv);

        // x_dbl = xc @ W_x^T     [2048 x 80, K=1536]
        gemm_f32_wmma<<<dim3((DT_RANK + 2 * D_STATE + 63) / 64, gy), gblk, 0, stream>>>(
            xc, D_INNER, W_x, xdbl, MROWS, DT_RANK + 2 * D_STATE, D_INNER, 0, LSEQ);

        // dt_lin = x_dbl[:, :48] @ W_dt^T   [2048 x 1536, K=48], lda=80
        gemm_f32_wmma<<<dim3((D_INNER + 63) / 64, gy), gblk, 0, stream>>>(
            xdbl, DT_RANK + 2 * D_STATE, W_dt, dtlin, MROWS, D_INNER, DT_RANK, 0, LSEQ);

        int tscan = NBATCH * D_INNER;
        scan_kernel<<<(tscan + 255) / 256, 256, 0, stream>>>(
            xz, xc, xdbl, dtlin, b_dt, A_log, Dp, ybuf);

        // y_out = y @ W_out^T    [2048 x 768, K=1536]
        gemm_f32_wmma<<<dim3((DIMX + 63) / 64, gy), gblk, 0, stream>>>(
            ybuf, D_INNER, W_out, yout, MROWS, DIMX, D_INNER, 0, LSEQ);
    }

    int tout = MROWS * DIMX;
    combine_kernel<<<(tout + 255) / 256, 256, 0, stream>>>(
        x, yo_fw, yo_bw, (float*)d_out, tout);
}